// Model_87754771792411
// MI455X (gfx1250) — compile-verified
//
#include <hip/hip_runtime.h>

// ---------------------------------------------------------------------------
// KGAT 3-layer forward for MI455X (gfx1250, wave32, WMMA).
//  - Aggregation: L2-resident f32 global atomics (node tables fit in 192MB L2;
//    the 3.2M-edge gather/scatter is the roofline, not the GEMMs).
//  - Bi-interaction GEMM: V_WMMA_F32_16X16X4_F32 (fp32 in / fp32 accum —
//    memory-bound workload, so keep full precision). One wave computes a
//    16x32 output strip (two 16x16 tiles) for BOTH paths, sharing the A
//    fragments across 4 WMMAs per K-step. Compile-time DIN/DOUT.
// ---------------------------------------------------------------------------

typedef __attribute__((ext_vector_type(2))) float v2f;
typedef __attribute__((ext_vector_type(8))) float v8f;

#define OUT_STRIDE 352   // 128 + 128 + 64 + 32

// ---------------------------------------------------------------- zero ws --
__global__ __launch_bounds__(256) void kgat_zero_f4(float4* __restrict__ p, long n4) {
  long i = (long)blockIdx.x * blockDim.x + threadIdx.x;
  long stride = (long)gridDim.x * blockDim.x;
  float4 z = make_float4(0.f, 0.f, 0.f, 0.f);
  for (; i < n4; i += stride) p[i] = z;
}

// -------------------------------------------- copy x into out[:, 0:128] ----
__global__ __launch_bounds__(256) void kgat_copy_x(const float* __restrict__ x,
                                                   float* __restrict__ out, int n) {
  long i = (long)blockIdx.x * blockDim.x + threadIdx.x;   // over n*32 float4s
  long total = (long)n * 32;
  if (i >= total) return;
  long row = i >> 5;
  int  c4  = (int)(i & 31);
  const float4* s = (const float4*)(x + row * 128) + c4;
  float4*       d = (float4*)(out + row * (long)OUT_STRIDE) + c4;
  *d = *s;
}

// -------------------------------- edge aggregation: h_n[dst] += a * x[src] --
// One float4 slice of one edge per thread; consecutive lanes cover consecutive
// slices of the same edge (coalesced gather). shift = log2(D/4).
__global__ __launch_bounds__(256) void kgat_aggregate(
    const float* __restrict__ x, int x_stride,
    const int* __restrict__ src, const int* __restrict__ dst,
    const float* __restrict__ attn, float* __restrict__ hn,
    int D, int shift, unsigned n_threads) {
  unsigned gid = blockIdx.x * blockDim.x + threadIdx.x;
  if (gid >= n_threads) return;
  unsigned e = gid >> shift;
  int off = (int)((gid & ((1u << shift) - 1u)) << 2);
  int s = src[e];
  int d = dst[e];
  float a = attn[e];
  const float4 xv = *(const float4*)(x + (size_t)s * x_stride + off);
  float* hp = hn + (size_t)d * D + off;
  __hip_atomic_fetch_add(hp + 0, a * xv.x, __ATOMIC_RELAXED, __HIP_MEMORY_SCOPE_AGENT);
  __hip_atomic_fetch_add(hp + 1, a * xv.y, __ATOMIC_RELAXED, __HIP_MEMORY_SCOPE_AGENT);
  __hip_atomic_fetch_add(hp + 2, a * xv.z, __ATOMIC_RELAXED, __HIP_MEMORY_SCOPE_AGENT);
  __hip_atomic_fetch_add(hp + 3, a * xv.w, __ATOMIC_RELAXED, __HIP_MEMORY_SCOPE_AGENT);
}

__device__ __forceinline__ float kgat_lrelu(float v) {
  return (v > 0.0f) ? v : 0.01f * v;               // jax.nn.leaky_relu default slope
}

// ---------------------------------------------- fused bi-interaction GEMM --
// out[:, col_off:col_off+DOUT] =
//   leaky_relu((x+hn)@W1 + b1) + leaky_relu((x*hn)@W2 + b2)
// One wave computes a 16(row) x 32(col) strip: two 16x16 tiles per path.
// V_WMMA_F32_16X16X4_F32 layouts (CDNA5 ISA), r=lane&15, h=lane>>4:
//   A 16x4 : VGPR j -> A[r][2h+j]   (contiguous pair -> one b64 load)
//   B 4x16 : VGPR j -> B[2h+j][r]
//   C 16x16: VGPR j -> C[j+8h][r]
template <int DIN, int DOUT>
__global__ __launch_bounds__(32) void kgat_bi_gemm(
    const float* __restrict__ xin, int x_stride,
    const float* __restrict__ hn,
    const float* __restrict__ W1, const float* __restrict__ b1,
    const float* __restrict__ W2, const float* __restrict__ b2,
    float* __restrict__ out, int col_off, int n_nodes) {
  const int tile_m = blockIdx.x * 16;
  const int tile_n = blockIdx.y * 32;              // two 16-wide column tiles
  const int lane = threadIdx.x;
  const int r = lane & 15;
  const int h = lane >> 4;

  int mr = tile_m + r;
  if (mr >= n_nodes) mr = n_nodes - 1;   // clamp loads; no early-out (EXEC all-1s for WMMA)
  const float* xrow = xin + (size_t)mr * x_stride;
  const float* hrow = hn + (size_t)mr * DIN;
  const float* w1a = W1 + tile_n + r;
  const float* w1b = w1a + 16;
  const float* w2a = W2 + tile_n + r;
  const float* w2b = w2a + 16;

  v8f acc1a = {}, acc1b = {};                      // (x+h)@W1 path, tiles a/b
  v8f acc2a = {}, acc2b = {};                      // (x*h)@W2 path, tiles a/b

#pragma unroll 4
  for (int k0 = 0; k0 < DIN; k0 += 4) {
    const int k = k0 + 2 * h;                      // 8B-aligned (k even)
    float2 xv = *(const float2*)(xrow + k);
    float2 gv = *(const float2*)(hrow + k);
    v2f a1; a1.x = xv.x + gv.x; a1.y = xv.y + gv.y;
    v2f a2; a2.x = xv.x * gv.x; a2.y = xv.y * gv.y;
    v2f f1a; f1a.x = w1a[k * DOUT]; f1a.y = w1a[(k + 1) * DOUT];
    v2f f1b; f1b.x = w1b[k * DOUT]; f1b.y = w1b[(k + 1) * DOUT];
    v2f f2a; f2a.x = w2a[k * DOUT]; f2a.y = w2a[(k + 1) * DOUT];
    v2f f2b; f2b.x = w2b[k * DOUT]; f2b.y = w2b[(k + 1) * DOUT];
    acc1a = __builtin_amdgcn_wmma_f32_16x16x4_f32(false, a1, false, f1a, (short)0,
                                                  acc1a, false, false);
    acc1b = __builtin_amdgcn_wmma_f32_16x16x4_f32(false, a1, false, f1b, (short)0,
                                                  acc1b, false, false);
    acc2a = __builtin_amdgcn_wmma_f32_16x16x4_f32(false, a2, false, f2a, (short)0,
                                                  acc2a, false, false);
    acc2b = __builtin_amdgcn_wmma_f32_16x16x4_f32(false, a2, false, f2b, (short)0,
                                                  acc2b, false, false);
  }

  const float bias1a = b1[tile_n + r];
  const float bias1b = b1[tile_n + 16 + r];
  const float bias2a = b2[tile_n + r];
  const float bias2b = b2[tile_n + 16 + r];
  float* obase = out + col_off + tile_n + r;
#pragma unroll
  for (int j = 0; j < 8; ++j) {
    int m = tile_m + j + 8 * h;
    if (m < n_nodes) {
      float va = kgat_lrelu(acc1a[j] + bias1a) + kgat_lrelu(acc2a[j] + bias2a);
      float vb = kgat_lrelu(acc1b[j] + bias1b) + kgat_lrelu(acc2b[j] + bias2b);
      obase[(size_t)m * OUT_STRIDE] = va;
      obase[(size_t)m * OUT_STRIDE + 16] = vb;
    }
  }
}

// ---------------------------------------------------------------------------
extern "C" void kernel_launch(void* const* d_in, const int* in_sizes, int n_in,
                              void* d_out, int out_size, void* d_ws, size_t ws_size,
                              hipStream_t stream) {
  const float* x    = (const float*)d_in[0];
  const int*   src  = (const int*)d_in[1];
  const int*   dst  = (const int*)d_in[2];
  const float* attn = (const float*)d_in[3];
  const float* W1[3] = {(const float*)d_in[4],  (const float*)d_in[8],  (const float*)d_in[12]};
  const float* b1[3] = {(const float*)d_in[5],  (const float*)d_in[9],  (const float*)d_in[13]};
  const float* W2[3] = {(const float*)d_in[6],  (const float*)d_in[10], (const float*)d_in[14]};
  const float* b2[3] = {(const float*)d_in[7],  (const float*)d_in[11], (const float*)d_in[15]};
  float* out = (float*)d_out;
  float* hn  = (float*)d_ws;                       // N*128 floats = 51.2 MB scratch

  const int N = in_sizes[0] / 128;                 // 100000
  const int E = in_sizes[1];                       // 3200000
  const int din[3]    = {128, 128, 64};
  const int coloff[4] = {0, 128, 256, 320};

  // out[:, 0:128] = x
  {
    long total = (long)N * 32;
    int blocks = (int)((total + 255) / 256);
    kgat_copy_x<<<blocks, 256, 0, stream>>>(x, out, N);
  }

  for (int l = 0; l < 3; ++l) {
    const int D = din[l];
    const float* xin = (l == 0) ? x : (out + coloff[l]);
    const int xstride = (l == 0) ? 128 : OUT_STRIDE;

    // zero h_n scratch (ws is poisoned by the harness; re-zero every call)
    {
      long n4 = (long)N * D / 4;
      kgat_zero_f4<<<2048, 256, 0, stream>>>((float4*)hn, n4);
    }

    // edge-weighted aggregation into h_n
    {
      const int shift = (D == 128) ? 5 : 4;        // log2(D/4) threads per edge
      unsigned n_threads = (unsigned)E << shift;
      int blocks = (int)((n_threads + 255u) / 256u);
      kgat_aggregate<<<blocks, 256, 0, stream>>>(xin, xstride, src, dst, attn,
                                                 hn, D, shift, n_threads);
    }

    // fused bi-interaction GEMM -> out[:, coloff[l+1] : +dout[l]]
    {
      const int mtiles = (N + 15) / 16;
      if (l == 0) {
        dim3 grid(mtiles, 128 / 32);
        kgat_bi_gemm<128, 128><<<grid, 32, 0, stream>>>(
            xin, xstride, hn, W1[l], b1[l], W2[l], b2[l], out, coloff[l + 1], N);
      } else if (l == 1) {
        dim3 grid(mtiles, 64 / 32);
        kgat_bi_gemm<128, 64><<<grid, 32, 0, stream>>>(
            xin, xstride, hn, W1[l], b1[l], W2[l], b2[l], out, coloff[l + 1], N);
      } else {
        dim3 grid(mtiles, 32 / 32);
        kgat_bi_gemm<64, 32><<<grid, 32, 0, stream>>>(
            xin, xstride, hn, W1[l], b1[l], W2[l], b2[l], out, coloff[l + 1], N);
      }
    }
  }
}